// NonlinearMPC_15006615733276
// MI455X (gfx1250) — compile-verified
//
#include <hip/hip_runtime.h>

// CDNA5 / gfx1250, wave32. fp32 WMMA (V_WMMA_F32_16X16X4_F32);
// async global->LDS staging (ASYNCcnt) for weights + u tiles; NT output stores.
// 512 threads = 16 wave32s: 1 GEMM1 tile/wave, 1 phase-B job/wave -> minimal
// per-step serial depth for the latency-bound 50-step recurrence.

typedef float v2f __attribute__((ext_vector_type(2)));
typedef float v8f __attribute__((ext_vector_type(8)));

#define C_NX  64
#define C_NU  16
#define C_KZ  80     // NX + NU
#define C_HID 256
#define C_H   50
#define BM    16     // batch rows per workgroup -> 256 workgroups
#define NT    512    // threads per block (16 waves)
#define ZS    81     // sZ row stride (odd -> conflict-free column-strided reads)
#define HS    257    // sH row stride

// Branchless fp32 tanh: one v_exp_f32, no EXEC divergence.
__device__ __forceinline__ float fast_tanh(float x) {
    float ax = __builtin_fabsf(x);
    float e  = __expf(-2.0f * ax);
    float t  = (1.0f - e) / (1.0f + e);
    return __builtin_copysignf(t, x);
}

// ------------------------------------------------------------------
// Kernel 0: control accumulation (only the last SQP iteration matters).
// ------------------------------------------------------------------
__global__ void prep_controls(const float* __restrict__ noise,
                              float* __restrict__ u_lin,
                              float* __restrict__ out0,
                              int batch, int nit) {
    long total = (long)batch * (C_H * C_NU);
    long idx = (long)blockIdx.x * blockDim.x + threadIdx.x;
    if (idx >= total) return;
    float acc = 0.f;
    for (int k = 0; k < nit - 1; ++k) acc += noise[(long)k * total + idx];
    u_lin[idx] = 0.001f * acc;
    int j = (int)(idx % (C_H * C_NU));
    if (j < C_NU) {
        float last = (nit >= 1) ? noise[(long)(nit - 1) * total + idx] : 0.f;
        long b = idx / (C_H * C_NU);
        out0[b * C_NU + j] = 0.001f * (acc + last);
    }
}

// ------------------------------------------------------------------
// Kernel 1: W1T[k,m] = W1[m,k];  s[k] = sum_n W2[k,n] (ones-cotangent VJP).
// ------------------------------------------------------------------
__global__ void prep_weights(const float* __restrict__ W1,
                             const float* __restrict__ W2,
                             float* __restrict__ W1T,
                             float* __restrict__ svec) {
    int t = blockIdx.x * blockDim.x + threadIdx.x;
    if (t < C_HID * C_KZ) {
        int k = t / C_KZ, m = t % C_KZ;
        W1T[t] = W1[m * C_HID + k];
    }
    if (t < C_HID) {
        float acc = 0.f;
        for (int n = 0; n < C_NX; ++n) acc += W2[t * C_NX + n];
        svec[t] = acc;
    }
}

// Async global->LDS copies (tracked by ASYNCcnt).
__device__ __forceinline__ void async_g2l_b32(uint32_t lds_off, const float* gp) {
    asm volatile("global_load_async_to_lds_b32 %0, %1, off"
                 :: "v"(lds_off), "v"(gp) : "memory");
}
__device__ __forceinline__ void async_g2l_b128(uint32_t lds_off, const float* gp) {
    asm volatile("global_load_async_to_lds_b128 %0, %1, off"
                 :: "v"(lds_off), "v"(gp) : "memory");
}
__device__ __forceinline__ void wait_async0() {
    asm volatile("s_wait_asynccnt 0x0" ::: "memory");
}

// ------------------------------------------------------------------
// Main kernel: persistent 50-step rollout per 16-row batch tile with fused
// linearization. Per step: GEMM1 | barrier | async u(t+1) | GEMM2/3/4 |
// wait_async | barrier.  W1T lives in LDS (stride 80 dwords = 16 mod 64
// banks -> the two lane-halves of each B-fragment hit disjoint banks).
// ------------------------------------------------------------------
__global__ __launch_bounds__(NT)
void mpc_rollout(const float* __restrict__ x0,
                 const float* __restrict__ W1,    // [80,256]
                 const float* __restrict__ b1,    // [256]
                 const float* __restrict__ W2,    // [256,64]
                 const float* __restrict__ W1Tg,  // [256,80]
                 const float* __restrict__ svec,  // [256]
                 const float* __restrict__ u_lin, // [B,800]
                 float* __restrict__ xtraj,       // [B,51,64]
                 float* __restrict__ xpred,       // [B,50,64]
                 float* __restrict__ gx,          // [B,50,64]
                 float* __restrict__ gu)          // [B,50,16]
{
    __shared__ float sZ[BM * ZS];                       //  5.2 KB state tile
    __shared__ float sH[BM * HS];                       // 16.4 KB tanh tile
    __shared__ __align__(16) float sW1T[C_HID * C_KZ];  // 80 KB W1 transposed
    __shared__ float sB1[C_HID];
    __shared__ float sS[C_HID];

    const int tid  = threadIdx.x;
    const int wv   = tid >> 5;        // 16 wave32s
    const int lane = tid & 31;
    const int r0   = blockIdx.x * BM;

    // u-tile async mapping: 16x16 = 256 dwords, one per thread of waves 0-7
    const int um0 = (tid & 255) >> 4;
    const int uc0 = tid & 15;
    const long ubase = (long)(r0 + um0) * (C_H * C_NU) + uc0;
    const bool uworker = tid < BM * C_NU;   // waves 0-7 (uniform per wave)

    if (tid < C_HID) { sB1[tid] = b1[tid]; sS[tid] = svec[tid]; }
    // stage x0 tile; emit x_traj[:,0]
    for (int i = tid; i < BM * C_NX; i += NT) {
        int m = i >> 6, c = i & 63;
        float v = x0[(long)(r0 + m) * C_NX + c];
        sZ[m * ZS + c] = v;
        __builtin_nontemporal_store(v, &xtraj[((long)(r0 + m) * (C_H + 1)) * C_NX + c]);
    }
    // async-stage W1T (20480 dwords as b128 chunks) and u(0)
    for (int i = tid * 4; i < C_HID * C_KZ; i += NT * 4)
        async_g2l_b128((uint32_t)(uintptr_t)&sW1T[i], W1Tg + i);
    if (uworker)
        async_g2l_b32((uint32_t)(uintptr_t)&sZ[um0 * ZS + 64 + uc0], u_lin + ubase);
    wait_async0();
    __syncthreads();

    const int lm = lane & 15;           // row(M)/col(N) within a 16x16 tile
    const int kk = (lane >> 4) * 2;     // lanes 0-15: K={0,1}; 16-31: K={2,3}
    const int rhalf = (lane >> 4) << 3; // C/D frag: lanes 16-31 are rows M=8..15

    for (int t = 0; t < C_H; ++t) {
        // ---- phase A: GEMM1  H = tanh(Z @ W1 + b1); 16 tiles, 1/wave ----
        {
            int col0 = wv * 16;
            v8f acc0 = {}, acc1 = {};
            for (int k0 = 0; k0 < C_KZ; k0 += 8) {
                v2f a0, b0, a1, b1f;
                a0.x  = sZ[lm * ZS + k0 + kk];
                a0.y  = sZ[lm * ZS + k0 + kk + 1];
                b0.x  = W1[(k0 + kk) * C_HID + col0 + lm];      // LICM-hoisted
                b0.y  = W1[(k0 + kk + 1) * C_HID + col0 + lm];
                a1.x  = sZ[lm * ZS + k0 + 4 + kk];
                a1.y  = sZ[lm * ZS + k0 + 4 + kk + 1];
                b1f.x = W1[(k0 + 4 + kk) * C_HID + col0 + lm];
                b1f.y = W1[(k0 + 5 + kk) * C_HID + col0 + lm];
                acc0 = __builtin_amdgcn_wmma_f32_16x16x4_f32(
                           false, a0, false, b0, (short)0, acc0, false, false);
                acc1 = __builtin_amdgcn_wmma_f32_16x16x4_f32(
                           false, a1, false, b1f, (short)0, acc1, false, false);
            }
            int colg = col0 + lm;
            float bb = sB1[colg];
            #pragma unroll
            for (int r = 0; r < 8; ++r)
                sH[(r + rhalf) * HS + colg] = fast_tanh(acc0[r] + acc1[r] + bb);
        }
        __syncthreads();

        // ---- async-issue u(t+1); hidden behind phase B ----
        if (t + 1 < C_H && uworker)
            async_g2l_b32((uint32_t)(uintptr_t)&sZ[um0 * ZS + 64 + uc0],
                          u_lin + ubase + (long)(t + 1) * C_NU);
        if (t + 2 < C_H && tid < BM)   // prefetch u(t+2) cachelines
            __builtin_prefetch(&u_lin[(long)(r0 + tid) * (C_H * C_NU) + (t + 2) * C_NU], 0, 1);

        // ---- phase B: 9 tile jobs (4 x_next, 4 gx, 1 gu), 1/wave ----
        if (wv < 9) {
            int job = wv;
            if (job < 4) {
                // x_next = x + 0.1 * H @ W2
                int col0 = job * 16;
                v8f acc0 = {}, acc1 = {};
                for (int k0 = 0; k0 < C_HID; k0 += 8) {
                    v2f a0, b0, a1, b1f;
                    a0.x  = sH[lm * HS + k0 + kk];
                    a0.y  = sH[lm * HS + k0 + kk + 1];
                    b0.x  = W2[(k0 + kk) * C_NX + col0 + lm];
                    b0.y  = W2[(k0 + kk + 1) * C_NX + col0 + lm];
                    a1.x  = sH[lm * HS + k0 + 4 + kk];
                    a1.y  = sH[lm * HS + k0 + 4 + kk + 1];
                    b1f.x = W2[(k0 + 4 + kk) * C_NX + col0 + lm];
                    b1f.y = W2[(k0 + 5 + kk) * C_NX + col0 + lm];
                    acc0 = __builtin_amdgcn_wmma_f32_16x16x4_f32(
                               false, a0, false, b0, (short)0, acc0, false, false);
                    acc1 = __builtin_amdgcn_wmma_f32_16x16x4_f32(
                               false, a1, false, b1f, (short)0, acc1, false, false);
                }
                int colg = col0 + lm;
                #pragma unroll
                for (int r = 0; r < 8; ++r) {
                    int row = r + rhalf;
                    float xn = sZ[row * ZS + colg] + 0.1f * (acc0[r] + acc1[r]);
                    sZ[row * ZS + colg] = xn;    // same-wave read->write, safe
                    long bg = r0 + row;
                    __builtin_nontemporal_store(
                        xn, &xtraj[(bg * (C_H + 1) + t + 1) * C_NX + colg]);
                    __builtin_nontemporal_store(
                        xn, &xpred[(bg * C_H + t) * C_NX + colg]);
                }
            } else if (job < 8) {
                // gx = 1 + Gh @ W1xT, Gh[m,k] = 0.1*s[k]*(1-h^2); W1T from LDS
                int col0 = (job - 4) * 16;
                v8f acc0 = {}, acc1 = {};
                for (int k0 = 0; k0 < C_HID; k0 += 8) {
                    v2f a0, b0, a1, b1f;
                    float h0 = sH[lm * HS + k0 + kk];
                    float h1 = sH[lm * HS + k0 + kk + 1];
                    float h2 = sH[lm * HS + k0 + 4 + kk];
                    float h3 = sH[lm * HS + k0 + 4 + kk + 1];
                    a0.x = 0.1f * sS[k0 + kk]     * (1.f - h0 * h0);
                    a0.y = 0.1f * sS[k0 + kk + 1] * (1.f - h1 * h1);
                    a1.x = 0.1f * sS[k0 + 4 + kk] * (1.f - h2 * h2);
                    a1.y = 0.1f * sS[k0 + 5 + kk] * (1.f - h3 * h3);
                    b0.x  = sW1T[(k0 + kk) * C_KZ + col0 + lm];
                    b0.y  = sW1T[(k0 + kk + 1) * C_KZ + col0 + lm];
                    b1f.x = sW1T[(k0 + 4 + kk) * C_KZ + col0 + lm];
                    b1f.y = sW1T[(k0 + 5 + kk) * C_KZ + col0 + lm];
                    acc0 = __builtin_amdgcn_wmma_f32_16x16x4_f32(
                               false, a0, false, b0, (short)0, acc0, false, false);
                    acc1 = __builtin_amdgcn_wmma_f32_16x16x4_f32(
                               false, a1, false, b1f, (short)0, acc1, false, false);
                }
                int colg = col0 + lm;
                #pragma unroll
                for (int r = 0; r < 8; ++r) {
                    long bg = r0 + r + rhalf;
                    __builtin_nontemporal_store(
                        1.0f + acc0[r] + acc1[r], &gx[(bg * C_H + t) * C_NX + colg]);
                }
            } else {
                // gu = Gh @ W1uT  (sW1T cols 64..79)
                v8f acc0 = {}, acc1 = {};
                for (int k0 = 0; k0 < C_HID; k0 += 8) {
                    v2f a0, b0, a1, b1f;
                    float h0 = sH[lm * HS + k0 + kk];
                    float h1 = sH[lm * HS + k0 + kk + 1];
                    float h2 = sH[lm * HS + k0 + 4 + kk];
                    float h3 = sH[lm * HS + k0 + 4 + kk + 1];
                    a0.x = 0.1f * sS[k0 + kk]     * (1.f - h0 * h0);
                    a0.y = 0.1f * sS[k0 + kk + 1] * (1.f - h1 * h1);
                    a1.x = 0.1f * sS[k0 + 4 + kk] * (1.f - h2 * h2);
                    a1.y = 0.1f * sS[k0 + 5 + kk] * (1.f - h3 * h3);
                    b0.x  = sW1T[(k0 + kk) * C_KZ + 64 + lm];
                    b0.y  = sW1T[(k0 + kk + 1) * C_KZ + 64 + lm];
                    b1f.x = sW1T[(k0 + 4 + kk) * C_KZ + 64 + lm];
                    b1f.y = sW1T[(k0 + 5 + kk) * C_KZ + 64 + lm];
                    acc0 = __builtin_amdgcn_wmma_f32_16x16x4_f32(
                               false, a0, false, b0, (short)0, acc0, false, false);
                    acc1 = __builtin_amdgcn_wmma_f32_16x16x4_f32(
                               false, a1, false, b1f, (short)0, acc1, false, false);
                }
                #pragma unroll
                for (int r = 0; r < 8; ++r) {
                    long bg = r0 + r + rhalf;
                    __builtin_nontemporal_store(
                        acc0[r] + acc1[r], &gu[(bg * C_H + t) * C_NU + lm]);
                }
            }
        }
        wait_async0();        // u(t+1) resident before the step barrier
        __syncthreads();
    }
}

// ------------------------------------------------------------------
extern "C" void kernel_launch(void* const* d_in, const int* in_sizes, int n_in,
                              void* d_out, int out_size, void* d_ws, size_t ws_size,
                              hipStream_t stream) {
    const float* x0    = (const float*)d_in[0];
    // d_in[1] = xref (zeros; unused by the outputs)
    const float* W1    = (const float*)d_in[2];
    const float* b1    = (const float*)d_in[3];
    const float* W2    = (const float*)d_in[4];
    const float* noise = (const float*)d_in[5];
    // d_in[6] = n_iterations (device scalar) — derived from sizes on host

    const int batch = in_sizes[0] / C_NX;
    const int nit   = in_sizes[5] / (batch * C_H * C_NU);

    float* out   = (float*)d_out;
    float* out0  = out;                                        // u10[:, :16]
    float* xtraj = out0  + (long)batch * C_NU;                 // [B,51,64]
    float* xpred = xtraj + (long)batch * (C_H + 1) * C_NX;     // [B,50,64]
    float* gx    = xpred + (long)batch * C_H * C_NX;           // [B,50,64]
    float* gu    = gx    + (long)batch * C_H * C_NX;           // [B,50,16]

    float* u_lin = (float*)d_ws;                               // [B,800]
    float* W1T   = u_lin + (long)batch * C_H * C_NU;           // [256,80]
    float* svec  = W1T + C_HID * C_KZ;                         // [256]

    {
        long total = (long)batch * C_H * C_NU;
        prep_controls<<<(int)((total + 255) / 256), 256, 0, stream>>>(
            noise, u_lin, out0, batch, nit);
    }
    prep_weights<<<(C_HID * C_KZ + 255) / 256, 256, 0, stream>>>(W1, W2, W1T, svec);
    mpc_rollout<<<batch / BM, NT, 0, stream>>>(x0, W1, b1, W2, W1T, svec, u_lin,
                                               xtraj, xpred, gx, gu);
}